// ExpAffineQuantizer_78202764525943
// MI455X (gfx1250) — compile-verified
//
#include <hip/hip_runtime.h>

// ---------------------------------------------------------------------------
// Log2-domain group fake-quantizer (G=128) for MI455X / gfx1250.
// One wave32 per group; async global->LDS double-buffered staging (ASYNCcnt);
// cross-lane min/max via shfl_xor; non-temporal b128 stores.
// HBM-bound: ~363 MB @ 23.3 TB/s => ~15.6 us target.
// ---------------------------------------------------------------------------

typedef float v4f __attribute__((ext_vector_type(4)));
typedef int   v4i __attribute__((vector_size(16)));   // matches builtin proto

#define WAVES_PER_BLOCK 8
#define GROUP 128
#define ITERS 4   // groups per wave

typedef __attribute__((address_space(1))) v4i as1_v4i;
typedef __attribute__((address_space(3))) v4i as3_v4i;

#if __has_builtin(__builtin_amdgcn_global_load_async_to_lds_b128)
#define HAVE_ASYNC_BUILTIN 1
#else
#define HAVE_ASYNC_BUILTIN 0
#endif

#if __has_builtin(__builtin_amdgcn_s_wait_asynccnt)
#define HAVE_WAIT_BUILTIN 1
#else
#define HAVE_WAIT_BUILTIN 0
#endif

__device__ __forceinline__ void async_load_group(const float* __restrict__ x,
                                                 long long g, float* lds_dst,
                                                 int lane) {
  const float* gsrc = x + (size_t)g * GROUP + (size_t)lane * 4;  // 16B per lane
  float* ldst = lds_dst + lane * 4;
#if HAVE_ASYNC_BUILTIN
  __builtin_amdgcn_global_load_async_to_lds_b128(
      (as1_v4i*)(size_t)gsrc,
      (as3_v4i*)(unsigned)(size_t)ldst,   // low 32 bits of flat addr = LDS offset
      0, 0);
#else
  unsigned ldsoff = (unsigned)(size_t)ldst;
  unsigned long long ga = (unsigned long long)(size_t)gsrc;
  asm volatile("global_load_async_to_lds_b128 %0, %1, off"
               :: "v"(ldsoff), "v"(ga) : "memory");
#endif
}

__device__ __forceinline__ void wait_async0() {
  asm volatile("" ::: "memory");
#if HAVE_WAIT_BUILTIN
  __builtin_amdgcn_s_wait_asynccnt(0);
#else
  asm volatile("s_wait_asynccnt 0" ::: "memory");
#endif
  asm volatile("" ::: "memory");
}

__device__ __forceinline__ void wait_async1() {
  asm volatile("" ::: "memory");
#if HAVE_WAIT_BUILTIN
  __builtin_amdgcn_s_wait_asynccnt(1);
#else
  asm volatile("s_wait_asynccnt 1" ::: "memory");
#endif
  asm volatile("" ::: "memory");
}

__device__ __forceinline__ float sigmoid_f(float v) {
  return 1.0f / (1.0f + __expf(-v));
}

__global__ __launch_bounds__(WAVES_PER_BLOCK * 32)
void logquant_g128_kernel(const float* __restrict__ x,
                          const float* __restrict__ upb,
                          const float* __restrict__ lowb,
                          float* __restrict__ out, int n_groups) {
  __shared__ __align__(16) float smem[WAVES_PER_BLOCK * 2 * GROUP];  // 8 KB

  const int lane = (int)(threadIdx.x & 31u);
  const int wave = (int)(threadIdx.x >> 5);
  const long long g0 =
      ((long long)blockIdx.x * WAVES_PER_BLOCK + wave) * ITERS;
  float* wbuf = smem + wave * (2 * GROUP);

  if (g0 < n_groups) async_load_group(x, g0, wbuf, lane);

#pragma unroll
  for (int i = 0; i < ITERS; ++i) {
    const long long g = g0 + i;
    const bool more = (i + 1 < ITERS) && (g + 1 < (long long)n_groups);
    if (more) async_load_group(x, g + 1, wbuf + ((i + 1) & 1) * GROUP, lane);

    // In-order async completion: allow the just-issued load to remain
    // outstanding, require the one we consume to be done.
    if (more) wait_async1(); else wait_async0();

    if (g < n_groups) {   // wave-uniform branch
      const v4f v = *(const v4f*)(wbuf + (i & 1) * GROUP + lane * 4);

      // group min / max: 4-wide lane-local, then 5-step wave32 butterfly
      float mn = fminf(fminf(v.x, v.y), fminf(v.z, v.w));
      float mx = fmaxf(fmaxf(v.x, v.y), fmaxf(v.z, v.w));
#pragma unroll
      for (int m = 16; m >= 1; m >>= 1) {
        mn = fminf(mn, __shfl_xor(mn, m, 32));
        mx = fmaxf(mx, __shfl_xor(mx, m, 32));
      }

      const float su = sigmoid_f(upb[g]);
      const float sl = sigmoid_f(lowb[g]);
      // |sig(up)*xmax| = sig(up)*|xmax| since sig>0
      float scale = fmaxf(su * fabsf(mx), sl * fabsf(mn));
      scale = fminf(fmaxf(scale, 1e-5f), 1e4f);
      const float inv = 1.0f / scale;

      v4f o;
#pragma unroll
      for (int c = 0; c < 4; ++c) {
        const float e = v[c];
        const float a = fabsf(e) * inv;        // abs modifier is free
        // Pre-clip of a to [1e-5,1e4] is redundant under the final [-7,0]
        // clamp of round(log2(a)) -- both saturate to the same codes.
        float r = rintf(__log2f(a));           // v_log_f32 + v_rndne_f32
        r = fminf(fmaxf(r, -7.0f), 0.0f);
        const float q = ldexpf(scale, (int)r); // exact 2^n * scale
        const float s = (e > 0.0f) ? 1.0f : ((e < 0.0f) ? -1.0f : 0.0f);
        o[c] = s * q;
      }
      // Output is never re-read by this kernel; NT store keeps the (L2-sized)
      // input resident across graph replays.
      __builtin_nontemporal_store(
          o, (v4f*)(out + (size_t)g * GROUP + (size_t)lane * 4));
    }
  }
}

extern "C" void kernel_launch(void* const* d_in, const int* in_sizes, int n_in,
                              void* d_out, int out_size, void* d_ws,
                              size_t ws_size, hipStream_t stream) {
  const float* x    = (const float*)d_in[0];
  const float* upb  = (const float*)d_in[1];
  const float* lowb = (const float*)d_in[2];
  float* out        = (float*)d_out;

  const int n_groups = in_sizes[1];  // upbound_factor has one entry per group
  if (n_groups <= 0) return;

  const int groups_per_block = WAVES_PER_BLOCK * ITERS;   // 32
  const int blocks = (n_groups + groups_per_block - 1) / groups_per_block;

  logquant_g128_kernel<<<blocks, WAVES_PER_BLOCK * 32, 0, stream>>>(
      x, upb, lowb, out, n_groups);
}